// APPNP_Model_85237920956921
// MI455X (gfx1250) — compile-verified
//
#include <hip/hip_runtime.h>

// APPNP GNN on MI455X (gfx1250, wave32, WGP).
// - Dense 128x128 GEMMs: V_WMMA_F32_16X16X4_F32 (exact fp32), weight matrix
//   staged once per workgroup into LDS (64KB of the WGP's 320KB) so the 8
//   waves share it instead of 8x re-reading L2.
// - APPNP: fused form x_{k+1} = 0.9*(A_hat x_k) + 0.1*h0. Q is pre-initialized
//   with the self-loop + teleport terms (replaces memset AND the blend pass),
//   edge weights prescaled by 0.9, P/Q ping-pong. Feature matrix (25.6MB) is
//   L2-resident (192MB L2) so the random edge gathers + f32 atomics run at L2
//   speed. Trailing ReLU is fused into the consumer GEMM's A-loads.

#define NNODES 50000
#define NEDGES 800000
#define DIN    128
#define FN     128
#define NOUT   16
#define KSTEPS 10
#define ALPHA  0.1f
#define BETA   0.9f

typedef __attribute__((ext_vector_type(2))) float v2f;
typedef __attribute__((ext_vector_type(8))) float v8f;

// ---------------- normalization ----------------
__global__ __launch_bounds__(256) void degree_kernel(const int* __restrict__ col,
                                                     float* __restrict__ deg) {
  int e = blockIdx.x * 256 + threadIdx.x;
  if (e < NEDGES) atomicAdd(&deg[col[e]], 1.0f);
}

__global__ __launch_bounds__(256) void dinv_kernel(float* __restrict__ deg) {
  int i = blockIdx.x * 256 + threadIdx.x;
  if (i < NNODES) deg[i] = rsqrtf(deg[i] + 1.0f);  // +1 = self loop; deg >= 1 always
}

// prescaled by (1-alpha) so the scatter directly produces 0.9 * A_hat * x
__global__ __launch_bounds__(256) void edgew_kernel(const int* __restrict__ row,
                                                    const int* __restrict__ col,
                                                    const float* __restrict__ dinv,
                                                    float* __restrict__ w) {
  int e = blockIdx.x * 256 + threadIdx.x;
  if (e < NEDGES) w[e] = BETA * dinv[row[e]] * dinv[col[e]];
}

// ---------------- dense GEMM (N x 128) @ (128 x 128) + bias, fp32 WMMA ----------------
// Block = 8 waves = one 16-row strip across all 128 cols. W staged in LDS once.
template <int RELU>
__global__ __launch_bounds__(256) void gemm128_wmma(const float* __restrict__ X,
                                                    const float* __restrict__ W,
                                                    const float* __restrict__ bias,
                                                    float* __restrict__ H0,
                                                    float* __restrict__ P) {
  __shared__ float ldsW[DIN * FN];  // 64 KB of the WGP's 320 KB LDS
  for (int i = threadIdx.x; i < (DIN * FN) / 4; i += 256)
    ((float4*)ldsW)[i] = ((const float4*)W)[i];
  __syncthreads();

  const int lane = threadIdx.x & 31;
  const int wave = threadIdx.x >> 5;
  const int tm   = blockIdx.x;          // [0, 3125)
  const int hi   = lane >> 4;
  const int lo   = lane & 15;
  const int m    = tm * 16 + lo;        // A-matrix row for this lane
  const int col  = wave * 16 + lo;      // B/C column
  const float* __restrict__ xr = X + (size_t)m * DIN;
  v8f acc = {};
#pragma unroll
  for (int k = 0; k < DIN; k += 4) {
    const int kk = k + 2 * hi;          // VGPR0 -> K=kk, VGPR1 -> K=kk+1
    v2f a, b;
    a.x = xr[kk];
    a.y = xr[kk + 1];
    if (RELU) { a.x = fmaxf(a.x, 0.0f); a.y = fmaxf(a.y, 0.0f); }
    b.x = ldsW[kk * FN + col];
    b.y = ldsW[(kk + 1) * FN + col];
    acc = __builtin_amdgcn_wmma_f32_16x16x4_f32(false, a, false, b, (short)0, acc,
                                                false, false);
  }
  const float bv = bias[col];
#pragma unroll
  for (int r = 0; r < 8; ++r) {
    const int   mr = tm * 16 + r + 8 * hi;   // C/D: VGPR r -> M = r + 8*hi
    const float v  = acc[r] + bv;
    H0[(size_t)mr * FN + col] = v;           // teleport anchor x0
    P [(size_t)mr * FN + col] = v;           // propagation state
  }
}

// ---------------- final projection (N x 128) @ (128 x 16) + bias, input ReLU ----------------
__global__ __launch_bounds__(256) void gemm_out_wmma(const float* __restrict__ X,
                                                     const float* __restrict__ Wc,
                                                     const float* __restrict__ bc,
                                                     float* __restrict__ logits) {
  __shared__ float ldsW[FN * NOUT];  // 8 KB
  for (int i = threadIdx.x; i < (FN * NOUT) / 4; i += 256)
    ((float4*)ldsW)[i] = ((const float4*)Wc)[i];
  __syncthreads();

  const int lane = threadIdx.x & 31;
  const int wave = threadIdx.x >> 5;
  const int tm   = blockIdx.x * 8 + wave;
  if (tm >= NNODES / 16) return;       // wave-uniform exit after barrier; EXEC all-1s
  const int hi = lane >> 4;
  const int lo = lane & 15;
  const int m  = tm * 16 + lo;
  const float* __restrict__ xr = X + (size_t)m * FN;
  v8f acc = {};
#pragma unroll
  for (int k = 0; k < FN; k += 4) {
    const int kk = k + 2 * hi;
    v2f a, b;
    a.x = fmaxf(xr[kk], 0.0f);         // fused ReLU on layer-3 output
    a.y = fmaxf(xr[kk + 1], 0.0f);
    b.x = ldsW[kk * NOUT + lo];
    b.y = ldsW[(kk + 1) * NOUT + lo];
    acc = __builtin_amdgcn_wmma_f32_16x16x4_f32(false, a, false, b, (short)0, acc,
                                                false, false);
  }
  const float bv = bc[lo];
#pragma unroll
  for (int r = 0; r < 8; ++r) {
    const int mr = tm * 16 + r + 8 * hi;
    logits[(size_t)mr * NOUT + lo] = acc[r] + bv;
  }
}

// ---------------- APPNP step init: Q = 0.9*dinv^2*P + 0.1*H0 ----------------
// Replaces both the memset of Q and the post-scatter blend pass.
__global__ __launch_bounds__(256) void appnp_init(const float4* __restrict__ P4,
                                                  const float4* __restrict__ H04,
                                                  const float* __restrict__ dinv,
                                                  float4* __restrict__ Q4) {
  const int t = blockIdx.x * 256 + threadIdx.x;     // over N*FN/4 = 1.6M float4s
  if (t >= NNODES * (FN / 4)) return;
  const int   i  = t >> 5;                          // node index (FN/4 == 32)
  const float di = dinv[i];
  const float s  = BETA * di * di;                  // 0.9 * self-loop weight
  const float4 pv = P4[t];
  const float4 hv = H04[t];
  float4 o;
  o.x = s * pv.x + ALPHA * hv.x;
  o.y = s * pv.y + ALPHA * hv.y;
  o.z = s * pv.z + ALPHA * hv.z;
  o.w = s * pv.w + ALPHA * hv.w;
  Q4[t] = o;
}

// ---------------- SpMM scatter: Q[col] += (0.9*w) * P[row] ----------------
// 32 lanes per edge, one float4 of features per lane -> 512B coalesced gather.
__global__ __launch_bounds__(256) void scatter_kernel(const int* __restrict__ row,
                                                      const int* __restrict__ col,
                                                      const float* __restrict__ w,
                                                      const float* __restrict__ P,
                                                      float* __restrict__ Q) {
  const long tid = (long)blockIdx.x * 256 + threadIdx.x;
  const int  e   = (int)(tid >> 5);
  if (e >= NEDGES) return;
  const int   lane = threadIdx.x & 31;
  const int   r    = row[e];          // wave-uniform
  const int   c    = col[e];
  const float we   = w[e];
  const float4 v = ((const float4*)(P + (size_t)r * FN))[lane];
  float* dst = Q + (size_t)c * FN + lane * 4;
  atomicAdd(dst + 0, we * v.x);
  atomicAdd(dst + 1, we * v.y);
  atomicAdd(dst + 2, we * v.z);
  atomicAdd(dst + 3, we * v.w);
}

// ---------------- log_softmax over 16 classes ----------------
__global__ __launch_bounds__(256) void logsoftmax_kernel(const float* __restrict__ logits,
                                                         float* __restrict__ out) {
  const int i = blockIdx.x * 256 + threadIdx.x;
  if (i >= NNODES) return;
  float v[NOUT];
  float m = -3.4e38f;
#pragma unroll
  for (int j = 0; j < NOUT; ++j) {
    v[j] = logits[(size_t)i * NOUT + j];
    m = fmaxf(m, v[j]);
  }
  float s = 0.0f;
#pragma unroll
  for (int j = 0; j < NOUT; ++j) s += expf(v[j] - m);
  const float ls = logf(s) + m;
#pragma unroll
  for (int j = 0; j < NOUT; ++j) out[(size_t)i * NOUT + j] = v[j] - ls;
}

extern "C" void kernel_launch(void* const* d_in, const int* in_sizes, int n_in,
                              void* d_out, int out_size, void* d_ws, size_t ws_size,
                              hipStream_t stream) {
  const float* x  = (const float*)d_in[0];
  const int*   ei = (const int*)d_in[1];
  const float* W1 = (const float*)d_in[2];
  const float* b1 = (const float*)d_in[3];
  const float* W2 = (const float*)d_in[4];
  const float* b2 = (const float*)d_in[5];
  const float* W3 = (const float*)d_in[6];
  const float* b3 = (const float*)d_in[7];
  const float* Wc = (const float*)d_in[8];
  const float* bc = (const float*)d_in[9];
  const int* row = ei;            // edge_index[0, :]
  const int* col = ei + NEDGES;   // edge_index[1, :]

  // workspace: dinv[N] | w[E] | B0 | B1 | B2 (each N*FN floats) ~= 77 MiB
  float* dinv = (float*)d_ws;
  float* wE   = dinv + NNODES;
  float* B0   = wE + NEDGES;
  float* B1   = B0 + (size_t)NNODES * FN;
  float* B2   = B1 + (size_t)NNODES * FN;

  // gcn_norm with self loops
  hipMemsetAsync(dinv, 0, NNODES * sizeof(float), stream);
  degree_kernel<<<(NEDGES + 255) / 256, 256, 0, stream>>>(col, dinv);
  dinv_kernel<<<(NNODES + 255) / 256, 256, 0, stream>>>(dinv);
  edgew_kernel<<<(NEDGES + 255) / 256, 256, 0, stream>>>(row, col, dinv, wE);

  auto appnp = [&](float* H0, float* P, float* Q) {
    float* p = P;
    float* q = Q;
    for (int k = 0; k < KSTEPS; ++k) {
      appnp_init<<<(NNODES * (FN / 4) + 255) / 256, 256, 0, stream>>>(
          (const float4*)p, (const float4*)H0, dinv, (float4*)q);
      scatter_kernel<<<((long)NEDGES * 32) / 256, 256, 0, stream>>>(row, col, wE, p, q);
      float* t = p; p = q; q = t;
    }
    // KSTEPS even -> result back in the original P buffer
  };

  // buffer rotation: layer input is only live during its GEMM -> becomes next Q
  gemm128_wmma<0><<<NNODES / 16, 256, 0, stream>>>(x, W1, b1, B0, B1);
  appnp(B0, B1, B2);                                     // result in B1 (pre-ReLU)
  gemm128_wmma<1><<<NNODES / 16, 256, 0, stream>>>(B1, W2, b2, B0, B2);
  appnp(B0, B2, B1);                                     // result in B2 (pre-ReLU)
  gemm128_wmma<1><<<NNODES / 16, 256, 0, stream>>>(B2, W3, b3, B0, B1);
  appnp(B0, B1, B2);                                     // result in B1 (pre-ReLU)

  gemm_out_wmma<<<(NNODES / 16 + 7) / 8, 256, 0, stream>>>(B1, Wc, bc, B2);
  logsoftmax_kernel<<<(NNODES + 255) / 256, 256, 0, stream>>>(B2, (float*)d_out);
}